// CustomConvLayer_13975823581360
// MI455X (gfx1250) — compile-verified
//
#include <hip/hip_runtime.h>
#include <hip/hip_bf16.h>

typedef __attribute__((ext_vector_type(2))) float v2f;
typedef __attribute__((ext_vector_type(4))) float v4f;
typedef __attribute__((ext_vector_type(8))) float v8f;

#define MSTRIDE 68   // LDS row pad: bank = (4*row + col) % 64 -> conflict-free b64 reads

// ---------------------------------------------------------------------------
// Kernel 1: fold d4 permutations + 7x7 kernel into a single 64x64 operator M.
// M[i][j] = (1/8) * sum_k C[i][k] * (sum_g d4[g][k][j])
// where C[(r,c),(a,b)] = ker[(a-r+3)*7 + (b-c+3)] when in [0,7).
// ---------------------------------------------------------------------------
__global__ void build_M_kernel(const float* __restrict__ d4,
                               const float* __restrict__ ker,
                               float* __restrict__ M) {
    __shared__ float T[64][64];
    __shared__ float K[49];
    const int tid = threadIdx.x;          // 64 threads
    if (tid < 49) K[tid] = ker[tid];
    for (int k = 0; k < 64; ++k) {
        float s = 0.f;
#pragma unroll
        for (int g = 0; g < 8; ++g) s += d4[g * 4096 + k * 64 + tid];
        T[k][tid] = s;                    // T[k][j=tid]
    }
    __syncthreads();
    const int r = tid >> 3, c = tid & 7;  // output position i = tid
    for (int j = 0; j < 64; ++j) {
        float acc = 0.f;
        for (int a = 0; a < 8; ++a) {
            const int kr = a - r + 3;
            if (kr < 0 || kr > 6) continue;
            for (int b = 0; b < 8; ++b) {
                const int kc = b - c + 3;
                if (kc < 0 || kc > 6) continue;
                acc += K[kr * 7 + kc] * T[a * 8 + b][j];
            }
        }
        M[tid * 64 + j] = 0.125f * acc;
    }
}

// ---------------------------------------------------------------------------
// Kernel 2: out[b,:] = M * x[b,:]  as WMMA GEMM.
// Per wave: grid-stride over 16-row batch chunks (~4 chunks/wave so the
// Msh fill + hoisted A-fragment ds loads amortize).
// D(16i x 16b) = A(M-tile 16x4) x B(x^T 4x16), K = 64 in 16 steps,
// 4 i-tiles -> 64 x v_wmma_f32_16x16x4_f32 per chunk.
// ---------------------------------------------------------------------------
__global__ void __launch_bounds__(256)
gemm_wmma_kernel(const float* __restrict__ x,
                 const float* __restrict__ M,
                 float* __restrict__ out,
                 int nchunks) {
    __shared__ __align__(16) float Msh[64 * MSTRIDE];
    for (int idx = threadIdx.x; idx < 64 * 64; idx += 256) {
        const int row = idx >> 6, col = idx & 63;
        Msh[row * MSTRIDE + col] = M[idx];
    }
    __syncthreads();

    const int lane = threadIdx.x & 31;
    const int wave = threadIdx.x >> 5;
    const int l15  = lane & 15;
    const int half = lane >> 4;           // 0: K={0,1}, 1: K={2,3} per WMMA frag

    const int waves_total = gridDim.x * 8;
    const int w0 = blockIdx.x * 8 + wave; // wave-uniform -> EXEC stays all-ones

    // A fragments: compiler hoists these 64 ds_load_b64 into persistent VGPRs,
    // loop-invariant across the chunk loop.
    const float* mBase = &Msh[l15 * MSTRIDE + 2 * half];

    for (int chunk = w0; chunk < nchunks; chunk += waves_total) {
        // B fragments: x^T tile. Lane holds x[base+l15][4*kk + 2*half + {0,1}].
        const float* xr = x + (size_t)(chunk * 16 + l15) * 64 + 2 * half;
        v2f bfrag[16];
#pragma unroll
        for (int kk = 0; kk < 16; ++kk)
            bfrag[kk] = *(const v2f*)(xr + 4 * kk);

        const v8f zero = {0.f, 0.f, 0.f, 0.f, 0.f, 0.f, 0.f, 0.f};
        v8f acc[4];
#pragma unroll
        for (int t = 0; t < 4; ++t) acc[t] = zero;

#pragma unroll
        for (int kk = 0; kk < 16; ++kk) {
#pragma unroll
            for (int t = 0; t < 4; ++t) {
                v2f afrag = *(const v2f*)(mBase + t * 16 * MSTRIDE + 4 * kk);
                acc[t] = __builtin_amdgcn_wmma_f32_16x16x4_f32(
                    /*neg_a=*/false, afrag, /*neg_b=*/false, bfrag[kk],
                    /*c_mod=*/(short)0, acc[t],
                    /*reuse_a=*/false, /*reuse_b=*/false);
            }
        }

        // D layout: lane holds 8 contiguous output columns i = t*16 + 8*half + v.
        float* orow = out + (size_t)(chunk * 16 + l15) * 64 + 8 * half;
#pragma unroll
        for (int t = 0; t < 4; ++t) {
            v4f lo = {acc[t][0], acc[t][1], acc[t][2], acc[t][3]};
            v4f hi = {acc[t][4], acc[t][5], acc[t][6], acc[t][7]};
            *(v4f*)(orow + t * 16)     = lo;
            *(v4f*)(orow + t * 16 + 4) = hi;
        }
    }
}

// Scalar tail for batch rows beyond the last full 16-row chunk (B % 16 != 0).
__global__ void tail_kernel(const float* __restrict__ x,
                            const float* __restrict__ M,
                            float* __restrict__ out,
                            int row0, int B) {
    const int gid = blockIdx.x * blockDim.x + threadIdx.x;
    const int b = row0 + (gid >> 6);
    if (b >= B) return;
    const int i = gid & 63;
    const float* xr = x + (size_t)b * 64;
    const float* mr = M + i * 64;
    float acc = 0.f;
#pragma unroll
    for (int j = 0; j < 64; ++j) acc += mr[j] * xr[j];
    out[(size_t)b * 64 + i] = acc;
}

extern "C" void kernel_launch(void* const* d_in, const int* in_sizes, int n_in,
                              void* d_out, int out_size, void* d_ws, size_t ws_size,
                              hipStream_t stream) {
    (void)n_in; (void)out_size; (void)ws_size;
    const float* x   = (const float*)d_in[0];   // [B, 8, 8, 1] fp32
    const float* d4  = (const float*)d_in[1];   // [8, 64, 64] fp32
    const float* ker = (const float*)d_in[2];   // [7, 7, 1, 1] fp32
    float* out = (float*)d_out;                 // [B, 8, 8, 1] fp32
    float* M   = (float*)d_ws;                  // 64*64 fp32 scratch

    const int B = in_sizes[0] / 64;

    build_M_kernel<<<1, 64, 0, stream>>>(d4, ker, M);

    const int nchunks = B / 16;
    if (nchunks > 0) {
        // ~4 chunks per wave: amortize Msh fill + hoisted A-fragment loads
        // while still launching enough waves (512 blocks -> 4096 waves for
        // the reference B=262144) to fill every SIMD.
        int blocks = (nchunks + 31) / 32;
        if (blocks > 65535) blocks = 65535;   // grid-stride handles the rest
        gemm_wmma_kernel<<<blocks, 256, 0, stream>>>(x, M, out, nchunks);
    }
    const int rem = B - nchunks * 16;
    if (rem > 0) {
        const int threads = rem * 64;
        tail_kernel<<<(threads + 255) / 256, 256, 0, stream>>>(x, M, out, nchunks * 16, B);
    }
}